// GNNLayerAttention_41686952575546
// MI455X (gfx1250) — compile-verified
//
#include <hip/hip_runtime.h>
#include <cstddef>

#define DIM 128
#define SLOPE 0.2f

typedef float v2f __attribute__((ext_vector_type(2)));
typedef float v8f __attribute__((ext_vector_type(8)));

__device__ __forceinline__ float leaky(float x) { return x >= 0.0f ? x : SLOPE * x; }

// ---------- 0. fold attention projection: u1 = Watt_w @ a1, u2 = Watt_w @ a2 ----------
__global__ void gat_prep(const float* __restrict__ Watt_w, const float* __restrict__ Watt_b,
                         const float* __restrict__ a, float* __restrict__ u1,
                         float* __restrict__ u2, float* __restrict__ c12, int* __restrict__ emax) {
    int k = threadIdx.x;  // 0..127
    float t1 = 0.0f, t2 = 0.0f;
    for (int d = 0; d < DIM; ++d) {
        float w = Watt_w[k * DIM + d];
        t1 = fmaf(w, a[d], t1);
        t2 = fmaf(w, a[DIM + d], t2);
    }
    u1[k] = t1;
    u2[k] = t2;
    if (k == 0) {
        float b1 = 0.0f, b2 = 0.0f;
        for (int d = 0; d < DIM; ++d) {
            b1 = fmaf(Watt_b[d], a[d], b1);
            b2 = fmaf(Watt_b[d], a[DIM + d], b2);
        }
        c12[0] = b1;
        c12[1] = b2;
        *emax = (int)0xFF800000;  // bit pattern of -inf
    }
}

// ---------- zero scratch (must run every call: atomics accumulate) ----------
__global__ void gat_zero(float* __restrict__ p, int n) {
    int i = blockIdx.x * blockDim.x + threadIdx.x;
    if (i < n) p[i] = 0.0f;
}

// ---------- 1. per-node attention scores: s = F[n]·u + c (one wave per node) ----------
__global__ void gat_node_scores(const float* __restrict__ feat, const float* __restrict__ u1,
                                const float* __restrict__ u2, const float* __restrict__ c12,
                                float* __restrict__ s1, float* __restrict__ s2, int N) {
    int wave = (blockIdx.x * blockDim.x + threadIdx.x) >> 5;
    int lane = threadIdx.x & 31;
    if (wave >= N) return;
    const float* f = feat + (size_t)wave * DIM;
    float a1 = 0.0f, a2 = 0.0f;
#pragma unroll
    for (int i = 0; i < DIM / 32; ++i) {
        int c = lane + 32 * i;
        float fv = f[c];
        a1 = fmaf(fv, u1[c], a1);
        a2 = fmaf(fv, u2[c], a2);
    }
#pragma unroll
    for (int m = 16; m; m >>= 1) {
        a1 += __shfl_xor(a1, m, 32);
        a2 += __shfl_xor(a2, m, 32);
    }
    if (lane == 0) {
        s1[wave] = a1 + c12[0];
        s2[wave] = a2 + c12[1];
    }
}

// ---------- 2. WMMA f32 GEMM: h_msg = F @ W1 + b1 ----------
// Requires M % 16 == 0 (true for this op: M = 50000). Two 16x16 M-tiles per
// wave share each B fragment: half the B traffic, two independent WMMA chains
// per k-step. Straight-line stores: no per-row guards.
__global__ void gat_gemm_msg(const float* __restrict__ A, const float* __restrict__ B,
                             const float* __restrict__ bias, float* __restrict__ C, int M) {
    int wave = (blockIdx.x * blockDim.x + threadIdx.x) >> 5;
    int lane = threadIdx.x & 31;
    int mTiles = M >> 4;
    int mPairs = (mTiles + 1) >> 1;
    int tiles = mPairs * (DIM / 16);
    if (wave >= tiles) return;
    int n0 = (wave & 7) << 4;
    int m0a = (wave >> 3) << 5;
    int m0b = m0a + 16;
    int half = lane >> 4, lr = lane & 15;

    int rma = m0a + lr;                   // always in range: m0a <= M-16
    int rmb = min(m0b + lr, M - 1);       // clamp odd-tile tail, EXEC stays all-ones
    const float* arow0 = A + (size_t)rma * DIM;
    const float* arow1 = A + (size_t)rmb * DIM;

    v8f acc0 = {}, acc1 = {};
#pragma unroll 4
    for (int kk = 0; kk < DIM / 4; ++kk) {
        int k0 = kk * 4 + half * 2;  // A/B VGPR0: K=0|K=2 per lane-half; VGPR1: K=1|K=3
        v2f bv;
        bv.x = B[(size_t)k0 * DIM + n0 + lr];
        bv.y = B[(size_t)(k0 + 1) * DIM + n0 + lr];
        v2f a0;
        a0.x = arow0[k0];
        a0.y = arow0[k0 + 1];
        v2f a1;
        a1.x = arow1[k0];
        a1.y = arow1[k0 + 1];
        acc0 = __builtin_amdgcn_wmma_f32_16x16x4_f32(false, a0, false, bv, (short)0, acc0, false, false);
        acc1 = __builtin_amdgcn_wmma_f32_16x16x4_f32(false, a1, false, bv, (short)0, acc1, false, false);
    }
    float bcol = bias[n0 + lr];
    float* c0 = C + (size_t)(m0a + 8 * half) * DIM + n0 + lr;
#pragma unroll
    for (int i = 0; i < 8; ++i) c0[(size_t)i * DIM] = acc0[i] + bcol;
    if (m0b < M) {  // uniform per wave in practice: only skips for the unpaired tail tile
        float* c1 = C + (size_t)(m0b + 8 * half) * DIM + n0 + lr;
#pragma unroll
        for (int i = 0; i < 8; ++i) c1[(size_t)i * DIM] = acc1[i] + bcol;
    }
}

// ---------- 3. global max of leaky(s1[src]+s2[dst]) ----------
__global__ void gat_edge_max(const int* __restrict__ src, const int* __restrict__ dst,
                             const float* __restrict__ s1, const float* __restrict__ s2,
                             int* __restrict__ emax, int E) {
    __shared__ float red[8];
    int tid = blockIdx.x * blockDim.x + threadIdx.x;
    int stride = gridDim.x * blockDim.x;
    float m = -__builtin_inff();
    for (int e = tid; e < E; e += stride) {
        float v = s1[src[e]] + s2[dst[e]];
        m = fmaxf(m, leaky(v));
    }
#pragma unroll
    for (int d = 16; d; d >>= 1) m = fmaxf(m, __shfl_xor(m, d, 32));
    if ((threadIdx.x & 31) == 0) red[threadIdx.x >> 5] = m;
    __syncthreads();
    if (threadIdx.x == 0) {
        float mm = red[0];
        for (int i = 1; i < 8; ++i) mm = fmaxf(mm, red[i]);
        // monotone float-max via signed-max (>=0) / unsigned-min (<0) on the same word
        if (mm >= 0.0f)
            atomicMax(emax, __float_as_int(mm));
        else
            atomicMin((unsigned int*)emax, (unsigned int)__float_as_int(mm));
    }
}

// ---------- 4. exp(e - max) and denominator segment sum ----------
__global__ void gat_edge_exp(const int* __restrict__ src, const int* __restrict__ dst,
                             const float* __restrict__ s1, const float* __restrict__ s2,
                             const int* __restrict__ emax, float* __restrict__ exp_e,
                             float* __restrict__ denom, int E) {
    int e = blockIdx.x * blockDim.x + threadIdx.x;
    if (e >= E) return;
    float mx = __int_as_float(*emax);
    float v = leaky(s1[src[e]] + s2[dst[e]]);
    float t = expf(v - mx);
    exp_e[e] = t;
    atomicAdd(&denom[dst[e]], t);
}

// ---------- 5. weighted message scatter (one wave per edge, f32 atomics) ----------
__global__ void gat_edge_scatter(const int* __restrict__ src, const int* __restrict__ dst,
                                 const float* __restrict__ exp_e, const float* __restrict__ denom,
                                 const float* __restrict__ h_msg, float* __restrict__ h_neigh, int E) {
    int wave = (blockIdx.x * blockDim.x + threadIdx.x) >> 5;
    int lane = threadIdx.x & 31;
    if (wave >= E) return;
    int s = src[wave], d = dst[wave];
    const float* hm = h_msg + (size_t)s * DIM;
    float* hn = h_neigh + (size_t)d * DIM;
    // overlap the random-row gather with the exp/denom dependency
    __builtin_prefetch(hm + lane, 0, 0);
    float w = exp_e[wave] / (denom[d] + 1e-9f);
#pragma unroll
    for (int i = 0; i < DIM / 32; ++i) {
        int c = lane + 32 * i;
        atomicAdd(&hn[c], hm[c] * w);
    }
}

// ---------- 6. WMMA GEMM with fused elementwise A and epilogue ----------
// out = leaky( F + h_neigh + (F .* h_neigh) @ W2 + b2 ).  Requires M % 16 == 0.
__global__ void gat_gemm_out(const float* __restrict__ feat, const float* __restrict__ hn,
                             const float* __restrict__ B, const float* __restrict__ bias,
                             float* __restrict__ out, int M) {
    int wave = (blockIdx.x * blockDim.x + threadIdx.x) >> 5;
    int lane = threadIdx.x & 31;
    int mTiles = M >> 4;
    int mPairs = (mTiles + 1) >> 1;
    int tiles = mPairs * (DIM / 16);
    if (wave >= tiles) return;
    int n0 = (wave & 7) << 4;
    int m0a = (wave >> 3) << 5;
    int m0b = m0a + 16;
    int half = lane >> 4, lr = lane & 15;

    int rma = m0a + lr;
    int rmb = min(m0b + lr, M - 1);
    const float* frow0 = feat + (size_t)rma * DIM;
    const float* hrow0 = hn + (size_t)rma * DIM;
    const float* frow1 = feat + (size_t)rmb * DIM;
    const float* hrow1 = hn + (size_t)rmb * DIM;

    v8f acc0 = {}, acc1 = {};
#pragma unroll 4
    for (int kk = 0; kk < DIM / 4; ++kk) {
        int k0 = kk * 4 + half * 2;
        v2f bv;
        bv.x = B[(size_t)k0 * DIM + n0 + lr];
        bv.y = B[(size_t)(k0 + 1) * DIM + n0 + lr];
        v2f a0;
        a0.x = frow0[k0] * hrow0[k0];
        a0.y = frow0[k0 + 1] * hrow0[k0 + 1];
        v2f a1;
        a1.x = frow1[k0] * hrow1[k0];
        a1.y = frow1[k0 + 1] * hrow1[k0 + 1];
        acc0 = __builtin_amdgcn_wmma_f32_16x16x4_f32(false, a0, false, bv, (short)0, acc0, false, false);
        acc1 = __builtin_amdgcn_wmma_f32_16x16x4_f32(false, a1, false, bv, (short)0, acc1, false, false);
    }
    float bcol = bias[n0 + lr];
    {
        size_t base = (size_t)(m0a + 8 * half) * DIM + n0 + lr;
#pragma unroll
        for (int i = 0; i < 8; ++i) {
            size_t idx = base + (size_t)i * DIM;
            out[idx] = leaky(feat[idx] + hn[idx] + acc0[i] + bcol);
        }
    }
    if (m0b < M) {
        size_t base = (size_t)(m0b + 8 * half) * DIM + n0 + lr;
#pragma unroll
        for (int i = 0; i < 8; ++i) {
            size_t idx = base + (size_t)i * DIM;
            out[idx] = leaky(feat[idx] + hn[idx] + acc1[i] + bcol);
        }
    }
}

extern "C" void kernel_launch(void* const* d_in, const int* in_sizes, int n_in,
                              void* d_out, int out_size, void* d_ws, size_t ws_size,
                              hipStream_t stream) {
    const int* indices = (const int*)d_in[0];
    const float* features = (const float*)d_in[1];
    const float* W1_w = (const float*)d_in[3];
    const float* W1_b = (const float*)d_in[4];
    const float* W2_w = (const float*)d_in[5];
    const float* W2_b = (const float*)d_in[6];
    const float* Watt_w = (const float*)d_in[7];
    const float* Watt_b = (const float*)d_in[8];
    const float* a = (const float*)d_in[9];
    float* out = (float*)d_out;

    int E = in_sizes[0] / 2;
    int N = in_sizes[1] / DIM;  // 50000: exact multiple of 16 (GEMM kernels rely on this)
    const int* src = indices;
    const int* dst = indices + E;

    // workspace layout (floats)
    float* ws = (float*)d_ws;
    size_t NF = (size_t)N * DIM;
    float* h_msg = ws;           // N*DIM
    float* h_neigh = ws + NF;    // N*DIM
    float* s1 = ws + 2 * NF;     // N
    float* s2 = s1 + N;          // N
    float* exp_e = s2 + N;       // E
    float* denom = exp_e + E;    // N
    float* u1 = denom + N;       // DIM
    float* u2 = u1 + DIM;        // DIM
    float* c12 = u2 + DIM;       // 2
    int* emax = (int*)(c12 + 2); // 1

    gat_prep<<<1, DIM, 0, stream>>>(Watt_w, Watt_b, a, u1, u2, c12, emax);

    int zn = (int)NF;
    gat_zero<<<(zn + 255) / 256, 256, 0, stream>>>(h_neigh, zn);
    gat_zero<<<(N + 255) / 256, 256, 0, stream>>>(denom, N);

    gat_node_scores<<<(N * 32 + 255) / 256, 256, 0, stream>>>(features, u1, u2, c12, s1, s2, N);

    int mTiles = N >> 4;
    int mPairs = (mTiles + 1) >> 1;
    int tiles = mPairs * (DIM / 16);
    int gBlocks = (tiles * 32 + 255) / 256;
    gat_gemm_msg<<<gBlocks, 256, 0, stream>>>(features, W1_w, W1_b, h_msg, N);

    gat_edge_max<<<1024, 256, 0, stream>>>(src, dst, s1, s2, emax, E);
    gat_edge_exp<<<(E + 255) / 256, 256, 0, stream>>>(src, dst, s1, s2, emax, exp_e, denom, E);

    long long scatterThreads = (long long)E * 32;
    int sBlocks = (int)((scatterThreads + 255) / 256);
    gat_edge_scatter<<<sBlocks, 256, 0, stream>>>(src, dst, exp_e, denom, h_msg, h_neigh, E);

    gat_gemm_out<<<gBlocks, 256, 0, stream>>>(features, h_neigh, W2_w, W2_b, out, N);
}